// ModelNew_5909874999962
// MI455X (gfx1250) — compile-verified
//
#include <hip/hip_runtime.h>

typedef float v2f __attribute__((ext_vector_type(2)));
typedef float v4f __attribute__((ext_vector_type(4)));
typedef float v8f __attribute__((ext_vector_type(8)));

#define NN   8192
#define BM   128
#define BN   128
#define BK   32
#define KP   (BK / 2)   // 16 k-pairs per tile
#define SA   144        // row stride (in float2) : 2*144 = 288 dwords = 32 mod 64
#define SB   144        //   -> lane halves read disjoint bank sets (conflict-free b64)

// C = tril( tril(A) @ tril(B) ), fp32, via V_WMMA_F32_16X16X4_F32.
// LDS holds k-pair-interleaved tiles so each WMMA fragment is ONE ds_load_b64:
//   As2[kp][m] = { A[m][2kp], A[m][2kp+1] }   (A transposed, pair-packed)
//   Bs2[kp][n] = { B[2kp][n], B[2kp+1][n] }   (B pair-packed)
// Triangle masking is applied only on boundary k-tiles (uniform branch).
__global__ __launch_bounds__(256)
void trimm_f32_wmma(const float* __restrict__ A,
                    const float* __restrict__ B,
                    float* __restrict__ C) {
  const int bn  = blockIdx.x;
  const int bm  = blockIdx.y;
  const int tid = threadIdx.x;
  const int RM  = bm * BM;
  const int RN  = bn * BN;

  if (bm < bn) {
    // Strictly upper-triangular block: output is exactly zero.
    v4f z = {0.f, 0.f, 0.f, 0.f};
#pragma unroll
    for (int i = 0; i < 16; ++i) {
      int idx = tid + i * 256;
      int row = idx >> 5;
      int cg  = (idx & 31) << 2;
      *(v4f*)(C + (size_t)(RM + row) * NN + (RN + cg)) = z;
    }
    return;
  }

  __shared__ v2f As2[KP][SA];
  __shared__ v2f Bs2[KP][SB];

  const int wave = tid >> 5;
  const int lane = tid & 31;
  const int half = lane >> 4;    // lane half selects K pair (ISA 16x4 f32 layout)
  const int lm   = lane & 15;

  const int wm = (wave & 3) * 32;   // 4x2 wave grid: 32 rows x 64 cols per wave
  const int wn = (wave >> 2) * 64;

  v8f acc[2][4];
#pragma unroll
  for (int tm = 0; tm < 2; ++tm)
#pragma unroll
    for (int tn = 0; tn < 4; ++tn)
      acc[tm][tn] = (v8f){0.f, 0.f, 0.f, 0.f, 0.f, 0.f, 0.f, 0.f};

  const int kEnd = RM + BM;           // nonzero band: k in [RN, RM+BM)
  for (int k0 = RN; k0 < kEnd; k0 += BK) {
    const bool needA = (k0 + BK - 1) > RM;        // A boundary tile (k > m possible)
    const bool needB = k0 < (RN + BM - 1);        // B boundary tile (n > k possible)

    // ---- Stage A tile: 128 rows x 32 k, transposed + k-pair packed ----
#pragma unroll
    for (int i = 0; i < 4; ++i) {
      int idx = tid + i * 256;        // 1024 float4 loads (coalesced along k)
      int row = idx >> 3;             // 0..127
      int kq  = idx & 7;              // k quad index
      int gm  = RM + row;
      int gk  = k0 + (kq << 2);
      v4f va = *(const v4f*)(A + (size_t)gm * NN + gk);
      if (needA) {
        va.x = (gk + 0 <= gm) ? va.x : 0.f;
        va.y = (gk + 1 <= gm) ? va.y : 0.f;
        va.z = (gk + 2 <= gm) ? va.z : 0.f;
        va.w = (gk + 3 <= gm) ? va.w : 0.f;
      }
      As2[kq * 2 + 0][row] = (v2f){va.x, va.y};
      As2[kq * 2 + 1][row] = (v2f){va.z, va.w};
    }
    // ---- Stage B tile: 32 k x 128 n, k-pair packed (2 adjacent k rows/thread) ----
#pragma unroll
    for (int i = 0; i < 2; ++i) {
      int idx = tid + i * 256;        // 0..511
      int p   = idx >> 5;             // k-pair 0..15
      int ng  = (idx & 31) << 2;      // 0,4,...,124
      int gk0 = k0 + 2 * p;
      int gn  = RN + ng;
      v4f b0 = *(const v4f*)(B + (size_t)gk0 * NN + gn);
      v4f b1 = *(const v4f*)(B + (size_t)(gk0 + 1) * NN + gn);
      if (needB) {
        b0.x = (gn + 0 <= gk0) ? b0.x : 0.f;
        b0.y = (gn + 1 <= gk0) ? b0.y : 0.f;
        b0.z = (gn + 2 <= gk0) ? b0.z : 0.f;
        b0.w = (gn + 3 <= gk0) ? b0.w : 0.f;
        b1.x = (gn + 0 <= gk0 + 1) ? b1.x : 0.f;
        b1.y = (gn + 1 <= gk0 + 1) ? b1.y : 0.f;
        b1.z = (gn + 2 <= gk0 + 1) ? b1.z : 0.f;
        b1.w = (gn + 3 <= gk0 + 1) ? b1.w : 0.f;
      }
      Bs2[p][ng + 0] = (v2f){b0.x, b1.x};
      Bs2[p][ng + 1] = (v2f){b0.y, b1.y};
      Bs2[p][ng + 2] = (v2f){b0.z, b1.z};
      Bs2[p][ng + 3] = (v2f){b0.w, b1.w};
    }
    __syncthreads();

    // ---- Compute: 8 k-steps of 4; each fragment is one ds_load_b64 ----
#pragma unroll
    for (int s = 0; s < 8; ++s) {
      const int kp = 2 * s + half;    // this lane-half's k-pair
      v2f a[2], b[4];
#pragma unroll
      for (int tm = 0; tm < 2; ++tm)
        a[tm] = As2[kp][wm + tm * 16 + lm];
#pragma unroll
      for (int tn = 0; tn < 4; ++tn)
        b[tn] = Bs2[kp][wn + tn * 16 + lm];
#pragma unroll
      for (int tm = 0; tm < 2; ++tm)
#pragma unroll
        for (int tn = 0; tn < 4; ++tn)
          acc[tm][tn] = __builtin_amdgcn_wmma_f32_16x16x4_f32(
              false, a[tm], false, b[tn],
              (short)0, acc[tm][tn], false, false);
    }
    __syncthreads();
  }

  // ---- Store: full store off-diagonal; diagonal blocks mask row >= col ----
  const bool diag = (bm == bn);
#pragma unroll
  for (int tm = 0; tm < 2; ++tm)
#pragma unroll
    for (int tn = 0; tn < 4; ++tn) {
#pragma unroll
      for (int r = 0; r < 8; ++r) {
        int grow = RM + wm + tm * 16 + r + 8 * half;  // D layout: vgpr r -> M=r / M=8+r
        int gcol = RN + wn + tn * 16 + lm;            // N = lane % 16
        float v = acc[tm][tn][r];
        if (diag && grow < gcol) v = 0.f;
        C[(size_t)grow * NN + gcol] = v;
      }
    }
}

extern "C" void kernel_launch(void* const* d_in, const int* in_sizes, int n_in,
                              void* d_out, int out_size, void* d_ws, size_t ws_size,
                              hipStream_t stream) {
  (void)in_sizes; (void)n_in; (void)out_size; (void)d_ws; (void)ws_size;
  const float* A = (const float*)d_in[0];
  const float* B = (const float*)d_in[1];
  float* C = (float*)d_out;
  dim3 grid(NN / BN, NN / BM);   // x = col block, y = row block
  trimm_f32_wmma<<<grid, 256, 0, stream>>>(A, B, C);
}